// GumbelQuantizer_62414464745614
// MI455X (gfx1250) — compile-verified
//
#include <hip/hip_runtime.h>
#include <math.h>

typedef __attribute__((ext_vector_type(2))) float v2f;
typedef __attribute__((ext_vector_type(8))) float v8f;

#define BT      65536          // B*T
#define NE      1024
#define ED      256
#define NB      32             // batch
#define TAUV    1.0f

#define MT      16             // rows per block tile
#define NWAVES  8
#define NPW     128            // columns per wave
#define A_STRIDE 260           // 16x256 z tile, padded (banks: m*4+k distinct)
#define L_STRIDE 1028          // 16x1024 logits tile, padded

#define ZQ_OFF   0
#define KL_OFF   (BT * ED)                 // 16777216
#define SAMP_OFF (KL_OFF + 1)
#define IND_OFF  (SAMP_OFF + NB * NE)

__global__ __launch_bounds__(256)
void gq_zero_kernel(float* __restrict__ out) {
    int i = blockIdx.x * blockDim.x + threadIdx.x;
    if (i < NB * NE) out[SAMP_OFF + i] = 0.0f;
}

__global__ __launch_bounds__(256)
void gq_main_kernel(const float* __restrict__ z,
                    const float* __restrict__ W,      // [NE, ED]
                    const float* __restrict__ bias,   // [NE]
                    const float* __restrict__ cb,     // [NE, ED]
                    const float* __restrict__ gn,     // [BT, NE]
                    float* __restrict__ out,
                    float* __restrict__ klpart)       // [gridDim.x]
{
    __shared__ float sA[MT * A_STRIDE];
    __shared__ float sL[MT * L_STRIDE];
    __shared__ float sKL[NWAVES];

    const int tid  = threadIdx.x;
    const int lane = tid & 31;
    const int wave = tid >> 5;
    const int r0   = blockIdx.x * MT;

    // ---- stage A tile (16 rows x 256 k) into LDS, coalesced ----
    for (int idx = tid; idx < MT * ED; idx += 256) {
        int m = idx >> 8;
        int k = idx & 255;
        sA[m * A_STRIDE + k] = z[(size_t)(r0 + m) * ED + k];
    }
    __syncthreads();

    // ---- GEMM: each wave computes 16 x 128 of logits via 8 WMMA tiles ----
    const int nlane = lane & 15;
    const int koff  = (lane >> 4) << 1;   // 0 or 2 (A/B K-split across half-waves)
    const int nbase = wave * NPW;

    v8f acc[8] = {};
    const float* sArow = &sA[nlane * A_STRIDE];   // M = lane%16

    for (int k0 = 0; k0 < ED; k0 += 4) {
        v2f a;
        a.x = sArow[k0 + koff];
        a.y = sArow[k0 + koff + 1];
        #pragma unroll
        for (int t = 0; t < 8; ++t) {
            const int n = nbase + t * 16 + nlane;
            const float* wp = &W[(size_t)n * ED + k0 + koff];
            v2f b;
            b.x = wp[0];
            b.y = wp[1];
            acc[t] = __builtin_amdgcn_wmma_f32_16x16x4_f32(
                false, a, false, b, (short)0, acc[t], false, false);
        }
    }

    // ---- epilogue: accumulators + bias -> LDS logits tile ----
    const int mbase = (lane >> 4) << 3;   // 0 or 8 (C layout half-wave M split)
    #pragma unroll
    for (int t = 0; t < 8; ++t) {
        const int ncol = nbase + t * 16 + nlane;
        const float bv = bias[ncol];
        #pragma unroll
        for (int r = 0; r < 8; ++r) {
            sL[(mbase + r) * L_STRIDE + ncol] = acc[t][r] + bv;
        }
    }
    __syncthreads();

    // ---- per-row softmax / argmax / KL: wave handles rows 2w, 2w+1 ----
    float klsum = 0.0f;
    for (int rr = 0; rr < 2; ++rr) {
        const int m = wave * 2 + rr;
        const int R = r0 + m;
        const float* gpr = &gn[(size_t)R * NE];

        float lv[32], tv[32];
        float m1 = -INFINITY, m2 = -INFINITY;
        int   bidx = lane;
        #pragma unroll
        for (int i = 0; i < 32; ++i) {
            const int c = lane + (i << 5);
            const float l = sL[m * L_STRIDE + c];
            const float t = (l + gpr[c]) * (1.0f / TAUV);
            lv[i] = l; tv[i] = t;
            if (t > m1) { m1 = t; bidx = c; }   // first occurrence within lane
            m2 = fmaxf(m2, l);
        }
        // wave32 reductions: max+argmax (ties -> smaller index), max
        #pragma unroll
        for (int off = 16; off > 0; off >>= 1) {
            float om1 = __shfl_xor(m1, off, 32);
            int   obi = __shfl_xor(bidx, off, 32);
            float om2 = __shfl_xor(m2, off, 32);
            if (om1 > m1 || (om1 == m1 && obi < bidx)) { m1 = om1; bidx = obi; }
            m2 = fmaxf(m2, om2);
        }
        // sums (from registers, no re-read of gnoise)
        float S1 = 0.0f, S2 = 0.0f;
        #pragma unroll
        for (int i = 0; i < 32; ++i) {
            S1 += expf(tv[i] - m1);
            S2 += expf(lv[i] - m2);
        }
        #pragma unroll
        for (int off = 16; off > 0; off >>= 1) {
            S1 += __shfl_xor(S1, off, 32);
            S2 += __shfl_xor(S2, off, 32);
        }
        // KL row: sum qy*log(qy + 1e-10)
        float kl = 0.0f;
        const float inv2 = 1.0f / S2;
        #pragma unroll
        for (int i = 0; i < 32; ++i) {
            const float q = expf(lv[i] - m2) * inv2;
            kl += q * logf(q + 1e-10f);
        }
        #pragma unroll
        for (int off = 16; off > 0; off >>= 1) kl += __shfl_xor(kl, off, 32);

        // straight-through coefficient: y_soft at argmax is exp(0)/S1
        const float s    = 1.0f / S1;
        const float coef = (1.0f - s) + s;

        // z_q row = coef * codebook[ind]
        const float* cbr = &cb[(size_t)bidx * ED];
        #pragma unroll
        for (int i = 0; i < 8; ++i) {
            const int j = lane + (i << 5);
            out[ZQ_OFF + (size_t)R * ED + j] = coef * cbr[j];
        }
        if (lane == 0) {
            out[IND_OFF + R] = (float)bidx;
            out[SAMP_OFF + bidx] = 1.0f;   // faithful flat scatter (idx < 1024)
            klsum += kl;
        }
    }

    if (lane == 0) sKL[wave] = klsum;
    __syncthreads();
    if (tid == 0) {
        float sum = 0.0f;
        #pragma unroll
        for (int w = 0; w < NWAVES; ++w) sum += sKL[w];
        klpart[blockIdx.x] = sum;
    }
}

__global__ __launch_bounds__(256)
void gq_kl_kernel(const float* __restrict__ klpart, float* __restrict__ out, int nparts) {
    __shared__ float red[256];
    float s = 0.0f;
    for (int i = threadIdx.x; i < nparts; i += 256) s += klpart[i];
    red[threadIdx.x] = s;
    __syncthreads();
    for (int off = 128; off > 0; off >>= 1) {
        if (threadIdx.x < off) red[threadIdx.x] += red[threadIdx.x + off];
        __syncthreads();
    }
    if (threadIdx.x == 0) out[KL_OFF] = red[0] * (1.0f / (float)BT);
}

extern "C" void kernel_launch(void* const* d_in, const int* in_sizes, int n_in,
                              void* d_out, int out_size, void* d_ws, size_t ws_size,
                              hipStream_t stream) {
    const float* z    = (const float*)d_in[0];   // [32,2048,256]
    const float* W    = (const float*)d_in[1];   // [1024,256]
    const float* bias = (const float*)d_in[2];   // [1024]
    const float* cb   = (const float*)d_in[3];   // [1024,256]
    const float* gn   = (const float*)d_in[4];   // [65536,1024]
    float* out = (float*)d_out;
    float* klpart = (float*)d_ws;                // 4096 floats of scratch

    const int nblocks = BT / MT;                 // 4096

    gq_zero_kernel<<<(NB * NE + 255) / 256, 256, 0, stream>>>(out);
    gq_main_kernel<<<nblocks, 256, 0, stream>>>(z, W, bias, cb, gn, out, klpart);
    gq_kl_kernel<<<1, 256, 0, stream>>>(klpart, out, nblocks);
}